// ScaledDotProductAttention4d_12558484374015
// MI455X (gfx1250) — compile-verified
//
#include <hip/hip_runtime.h>

typedef float v2f __attribute__((ext_vector_type(2)));
typedef float v8f __attribute__((ext_vector_type(8)));

#define WMMA_F32(a, b, c) \
    __builtin_amdgcn_wmma_f32_16x16x4_f32(false, (a), false, (b), (short)0, (c), false, false)

constexpr int Bn = 4, H = 8, T = 64, E = 256, D = 64;
constexpr int SLABS = Bn * H * T;                    // 2048
constexpr size_t OUT_ELEMS = (size_t)SLABS * E * D;  // 33,554,432 floats
constexpr float INV_TEMP = 0.125f;                   // 1/sqrt(d_k) = 1/8
constexpr float NEG_INF_V = -1e30f;
constexpr int LDS_STRIDE = 260;                      // 256 + 4 pad -> conflict-free A-frag reads

__global__ __launch_bounds__(32)
void sdpa4d_wmma_kernel(const float* __restrict__ q,
                        const float* __restrict__ k,
                        const float* __restrict__ v,
                        const int*   __restrict__ mask,
                        float* __restrict__ out,
                        float* __restrict__ attn)
{
    __shared__ float lds[16 * LDS_STRIDE];

    const int bid  = blockIdx.x;
    const int slab = bid >> 4;        // (b,h,t) index
    const int rb   = bid & 15;        // 16-row block within E
    const int lane = threadIdx.x;
    const int lo   = lane & 15;
    const int hi   = lane >> 4;

    const float* Q = q + (size_t)slab * E * D;
    const float* K = k + (size_t)slab * E * D;
    const float* V = v + (size_t)slab * E * D;

    const int bb = slab / (H * T);
    const int tt = slab % T;
    const int* M = mask + (((size_t)bb * T + tt) * E + (size_t)rb * 16) * E;

    // ---------------- scores = (Q/8) @ K^T : 16 x 256 row-block ----------------
    v8f acc[16];
    {
        const float* qrow = Q + (size_t)(rb * 16 + lo) * D + 2 * hi;
        v2f a[16];
#pragma unroll
        for (int kc = 0; kc < 16; ++kc)
            a[kc] = (*(const v2f*)(qrow + kc * 4)) * INV_TEMP;

#pragma unroll
        for (int ft = 0; ft < 16; ++ft) {
            const float* krow = K + (size_t)(ft * 16 + lo) * D + 2 * hi;
            // Batch all 16 B-fragment loads first -> 16 in-flight global_load_b64,
            // then drain with WMMAs (waits at decreasing loadcnt, not 0 each time).
            v2f b[16];
#pragma unroll
            for (int kc = 0; kc < 16; ++kc)
                b[kc] = *(const v2f*)(krow + kc * 4);
            v8f c = {};
#pragma unroll
            for (int kc = 0; kc < 16; ++kc)
                c = WMMA_F32(a[kc], b[kc], c);
            acc[ft] = c;
        }
    }

    // ------------- mask (streaming, non-temporal): (mask==0) ? -1e30 -------------
#pragma unroll
    for (int ft = 0; ft < 16; ++ft) {
#pragma unroll
        for (int r = 0; r < 8; ++r) {
            const int mv = __builtin_nontemporal_load(M + (r + 8 * hi) * E + ft * 16 + lo);
            if (mv == 0) acc[ft][r] = NEG_INF_V;
        }
    }

    // ---------------- row softmax over 256 columns ----------------
    float rmax[8];
#pragma unroll
    for (int r = 0; r < 8; ++r) {
        float m = acc[0][r];
#pragma unroll
        for (int ft = 1; ft < 16; ++ft) m = fmaxf(m, acc[ft][r]);
#pragma unroll
        for (int s = 1; s < 16; s <<= 1) m = fmaxf(m, __shfl_xor(m, s, 32));
        rmax[r] = m;
    }
    float inv_sum[8];
#pragma unroll
    for (int r = 0; r < 8; ++r) {
        float s = 0.0f;
#pragma unroll
        for (int ft = 0; ft < 16; ++ft) {
            const float e = __expf(acc[ft][r] - rmax[r]);
            acc[ft][r] = e;
            s += e;
        }
#pragma unroll
        for (int st = 1; st < 16; st <<= 1) s += __shfl_xor(s, st, 32);
        inv_sum[r] = 1.0f / s;
    }

    // ------- normalize, emit attn (non-temporal stream), stage block in LDS -------
    float* attn_ptr = attn + (size_t)slab * E * E + (size_t)rb * 16 * E;
#pragma unroll
    for (int ft = 0; ft < 16; ++ft) {
#pragma unroll
        for (int r = 0; r < 8; ++r) {
            const float p = acc[ft][r] * inv_sum[r];
            acc[ft][r] = p;
            __builtin_nontemporal_store(p, attn_ptr + (r + 8 * hi) * E + ft * 16 + lo);
            lds[(r + 8 * hi) * LDS_STRIDE + ft * 16 + lo] = p;
        }
    }
    __syncthreads();   // single-wave block; DS ops are in-order anyway

    // ---------------- out = attn @ V : (16x256) x (256x64) ----------------
    const float* vcol = V + 2 * hi * D;   // rows offset by 2*hi handled via f0 below
    float* out_ptr = out + (size_t)slab * E * D + (size_t)rb * 16 * D;
#pragma unroll
    for (int dt = 0; dt < 4; ++dt) {
        v8f c = {};
        const int col = dt * 16 + lo;
#pragma unroll
        for (int ft = 0; ft < 16; ++ft) {
            v2f aa[4], bbv[4];
#pragma unroll
            for (int kc = 0; kc < 4; ++kc) {
                const int f0 = ft * 16 + kc * 4 + 2 * hi;
                aa[kc] = *(const v2f*)(&lds[lo * LDS_STRIDE + f0]);  // ds_load_b64
                bbv[kc].x = V[(size_t)(f0 + 0) * D + col];
                bbv[kc].y = V[(size_t)(f0 + 1) * D + col];
            }
#pragma unroll
            for (int kc = 0; kc < 4; ++kc)
                c = WMMA_F32(aa[kc], bbv[kc], c);
        }
#pragma unroll
        for (int r = 0; r < 8; ++r)
            __builtin_nontemporal_store(c[r], out_ptr + (r + 8 * hi) * D + col);
    }
    (void)vcol;
}

extern "C" void kernel_launch(void* const* d_in, const int* in_sizes, int n_in,
                              void* d_out, int out_size, void* d_ws, size_t ws_size,
                              hipStream_t stream) {
    (void)in_sizes; (void)n_in; (void)d_ws; (void)ws_size; (void)out_size;
    const float* q    = (const float*)d_in[0];
    const float* k    = (const float*)d_in[1];
    const float* v    = (const float*)d_in[2];
    const int*   mask = (const int*)d_in[3];
    float* out  = (float*)d_out;               // 33,554,432 floats
    float* attn = (float*)d_out + OUT_ELEMS;   // 134,217,728 floats

    dim3 grid(SLABS * 16);   // 32768 waves: one per 16-row score block
    dim3 block(32);          // one wave32
    sdpa4d_wmma_kernel<<<grid, block, 0, stream>>>(q, k, v, mask, out, attn);
}